// SuperPointMatching_75574244540607
// MI455X (gfx1250) — compile-verified
//
#include <hip/hip_runtime.h>
#include <hip/hip_bf16.h>
#include <stdint.h>

#define NN 4096          // N == M
#define DD 256           // feature dim
#define KSEL 256         // top-k
#define HBINS 16384      // histogram bins over float bits [30:17]
#define CAND_MAX 16384   // candidate cap (must be power of 2 for bitonic)

typedef __attribute__((ext_vector_type(2))) float v2f;
typedef __attribute__((ext_vector_type(4))) float v4f;
typedef __attribute__((ext_vector_type(8))) float v8f;

// ---------------------------------------------------------------------------
// Kernel 1: S = exp(2*ref@src^T - 2), accumulate row sums / col sums.
// Grid (32,32), 256 threads (8 waves). Block computes a 128x128 tile:
//  - B panel (128 cols x 256 dims, 128KB) staged once into LDS with
//    global_load_async_to_lds_b128 (ASYNCcnt) and shared by all 8 waves.
//  - Each wave: 16x128 strip via V_WMMA_F32_16X16X4_F32 (full f32 precision).
//  - K-permutation trick: half-wave 0 feeds physical k..k+3, half-wave 1
//    k+4..k+7 (one b128 each); summation order change only, result identical.
// ---------------------------------------------------------------------------
__global__ __launch_bounds__(256)
void gemm_exp_kernel(const float* __restrict__ ref, const float* __restrict__ src,
                     float* __restrict__ S, float* __restrict__ rowsum,
                     float* __restrict__ colsum)
{
    extern __shared__ float bsh[];                     // [128 cols][256 dims]
    const int tid  = threadIdx.x;
    const int lane = tid & 31;
    const int wave = tid >> 5;                         // 0..7
    const int rowBase = blockIdx.y * 128 + wave * 16;
    const int colBase = blockIdx.x * 128;
    const int half = lane >> 4;                        // 0 | 1
    const int l15  = lane & 15;

    // ---- async-stage the B panel: 128KB = 8192 x 16B chunks, 32 per thread
    const uint32_t lds_base = (uint32_t)(uintptr_t)bsh;
    #pragma unroll
    for (int i = 0; i < 32; ++i) {
        const int c     = tid + i * 256;               // chunk id
        const int col   = c >> 6;                      // 64 chunks per column
        const int off16 = c & 63;
        const unsigned long long g =
            (unsigned long long)(uintptr_t)(src + (size_t)(colBase + col) * DD
                                            + (size_t)off16 * 4);
        const uint32_t l = lds_base + (uint32_t)c * 16u;
        asm volatile("global_load_async_to_lds_b128 %0, %1, off"
                     :: "v"(l), "v"(g) : "memory");
    }
    asm volatile("s_wait_asynccnt 0x0" ::: "memory");
    __syncthreads();

    // A fragment source: lane half 0 -> physical k..k+3, half 1 -> k+4..k+7
    const float* aptr = ref + (size_t)(rowBase + l15) * DD + half * 4;
    // B fragment source (LDS): same K assignment, col = t*16 + l15
    const float* bl = bsh + (size_t)l15 * DD + half * 4;

    v8f acc[8];
    #pragma unroll
    for (int t = 0; t < 8; ++t) acc[t] = (v8f){0.f,0.f,0.f,0.f,0.f,0.f,0.f,0.f};

    for (int k = 0; k < DD; k += 8) {
        v4f a4 = *(const v4f*)(aptr + k);
        v2f a0, a1;
        a0.x = a4.x; a0.y = a4.y; a1.x = a4.z; a1.y = a4.w;
        #pragma unroll
        for (int t = 0; t < 8; ++t) {
            v4f b4 = *(const v4f*)(bl + (size_t)t * 16 * DD + k);  // ds_load_b128
            v2f b0, b1;
            b0.x = b4.x; b0.y = b4.y; b1.x = b4.z; b1.y = b4.w;
            acc[t] = __builtin_amdgcn_wmma_f32_16x16x4_f32(
                false, a0, false, b0, (short)0, acc[t], false, false);
            acc[t] = __builtin_amdgcn_wmma_f32_16x16x4_f32(
                false, a1, false, b1, (short)0, acc[t], false, false);
        }
    }

    // C/D layout: VGPR e: lanes 0-15 -> (M=e, N=lane); lanes 16-31 -> (M=8+e)
    float rsum_part[8];
    #pragma unroll
    for (int e = 0; e < 8; ++e) rsum_part[e] = 0.f;

    #pragma unroll
    for (int t = 0; t < 8; ++t) {
        const int c = colBase + t * 16 + l15;
        float csum_part = 0.f;
        #pragma unroll
        for (int e = 0; e < 8; ++e) {
            float s = expf(2.0f * acc[t][e] - 2.0f);   // exp(-(2-2*dot))
            rsum_part[e] += s;
            csum_part    += s;
            const int r = rowBase + half * 8 + e;
            S[(size_t)r * NN + c] = s;
        }
        // combine the two row-halves (lane L <-> L+16 share the same column)
        csum_part += __shfl_xor(csum_part, 16, 32);
        if (half == 0) atomicAdd(&colsum[c], csum_part);
    }
    // reduce over the 16 columns held by each half-wave
    #pragma unroll
    for (int e = 0; e < 8; ++e) {
        float v = rsum_part[e];
        v += __shfl_xor(v, 1, 32);
        v += __shfl_xor(v, 2, 32);
        v += __shfl_xor(v, 4, 32);
        v += __shfl_xor(v, 8, 32);
        if (l15 == 0) atomicAdd(&rowsum[rowBase + half * 8 + e], v);
    }
}

// ---------------------------------------------------------------------------
// Kernel 2: reciprocals of the dual-normalization sums
// ---------------------------------------------------------------------------
__global__ void invert_kernel(const float* __restrict__ rowsum,
                              const float* __restrict__ colsum,
                              float* __restrict__ rinv, float* __restrict__ cinv)
{
    int i = blockIdx.x * blockDim.x + threadIdx.x;
    if (i < NN) {
        rinv[i] = 1.0f / rowsum[i];
        cinv[i] = 1.0f / colsum[i];
    }
}

// ---------------------------------------------------------------------------
// Kernel 3: histogram of final-score float bits (F > 0 => bit order == order)
// ---------------------------------------------------------------------------
__global__ __launch_bounds__(256)
void hist_kernel(const float* __restrict__ S, const float* __restrict__ rinv,
                 const float* __restrict__ cinv, unsigned int* __restrict__ ghist)
{
    extern __shared__ unsigned int lhist[];
    for (int i = threadIdx.x; i < HBINS; i += blockDim.x) lhist[i] = 0;
    __syncthreads();

    const size_t total = (size_t)NN * NN;
    for (size_t idx = (size_t)blockIdx.x * blockDim.x + threadIdx.x;
         idx < total; idx += (size_t)gridDim.x * blockDim.x) {
        const int i = (int)(idx >> 12);
        const int j = (int)(idx & (NN - 1));
        const float s = S[idx];
        const float f = s * s * rinv[i] * cinv[j];
        atomicAdd(&lhist[__float_as_uint(f) >> 17], 1u);
    }
    __syncthreads();
    for (int i = threadIdx.x; i < HBINS; i += blockDim.x) {
        unsigned int v = lhist[i];
        if (v) atomicAdd(&ghist[i], v);
    }
}

// ---------------------------------------------------------------------------
// Kernel 4: suffix scan from the top bin -> threshold bits; reset counter
// ctrl[0] = threshold bits, ctrl[1] = candidate counter (reset to 0)
// ---------------------------------------------------------------------------
__global__ void scan_kernel(const unsigned int* __restrict__ ghist,
                            unsigned int* __restrict__ ctrl)
{
    extern __shared__ unsigned int h[];
    for (int i = threadIdx.x; i < HBINS; i += blockDim.x) h[i] = ghist[i];
    __syncthreads();
    if (threadIdx.x == 0) {
        unsigned int cnt = 0;
        int b = HBINS - 1;
        for (; b > 0; --b) { cnt += h[b]; if (cnt >= KSEL) break; }
        ctrl[0] = ((unsigned int)b) << 17;   // lower bound of threshold bin
        ctrl[1] = 0;                         // candidate counter
    }
}

// ---------------------------------------------------------------------------
// Kernel 5: collect every element with bits >= threshold as packed candidates
// key = (bits << 32) | ~idx   (desc sort => value desc, ties: lower idx first)
// ---------------------------------------------------------------------------
__global__ __launch_bounds__(256)
void collect_kernel(const float* __restrict__ S, const float* __restrict__ rinv,
                    const float* __restrict__ cinv,
                    unsigned int* __restrict__ ctrl,
                    unsigned long long* __restrict__ cand)
{
    const unsigned int tb = ctrl[0];
    const size_t total = (size_t)NN * NN;
    for (size_t idx = (size_t)blockIdx.x * blockDim.x + threadIdx.x;
         idx < total; idx += (size_t)gridDim.x * blockDim.x) {
        const int i = (int)(idx >> 12);
        const int j = (int)(idx & (NN - 1));
        const float s = S[idx];
        const float f = s * s * rinv[i] * cinv[j];
        const unsigned int bits = __float_as_uint(f);
        if (bits >= tb) {
            unsigned int pos = atomicAdd(&ctrl[1], 1u);
            if (pos < CAND_MAX)
                cand[pos] = ((unsigned long long)bits << 32) |
                            (unsigned long long)(~(unsigned int)idx);
        }
    }
}

// ---------------------------------------------------------------------------
// Kernel 6: one workgroup bitonic sort (descending) of candidates in LDS,
// emit top-256 (ref_idx, src_idx, score). 16384 keys * 8B = 128KB dynamic LDS.
// ---------------------------------------------------------------------------
__global__ __launch_bounds__(1024)
void select_kernel(const unsigned long long* __restrict__ cand,
                   const unsigned int* __restrict__ ctrl,
                   int* __restrict__ ref_out, int* __restrict__ src_out,
                   float* __restrict__ score_out)
{
    extern __shared__ unsigned long long key[];
    unsigned int n = ctrl[1];
    if (n > CAND_MAX) n = CAND_MAX;
    for (int i = threadIdx.x; i < CAND_MAX; i += blockDim.x)
        key[i] = (i < (int)n) ? cand[i] : 0ull;
    __syncthreads();

    for (int size = 2; size <= CAND_MAX; size <<= 1) {
        for (int stride = size >> 1; stride > 0; stride >>= 1) {
            for (int t = threadIdx.x; t < CAND_MAX / 2; t += blockDim.x) {
                const int lo = ((t & ~(stride - 1)) << 1) | (t & (stride - 1));
                const int hi = lo + stride;
                const bool desc = ((lo & size) == 0);
                unsigned long long a = key[lo], b = key[hi];
                if (desc ? (a < b) : (a > b)) { key[lo] = b; key[hi] = a; }
            }
            __syncthreads();
        }
    }

    for (int t = threadIdx.x; t < KSEL; t += blockDim.x) {
        const unsigned long long kv = key[t];
        const unsigned int bits = (unsigned int)(kv >> 32);
        const unsigned int idx  = ~((unsigned int)kv);
        ref_out[t]   = (int)(idx >> 12);       // idx / 4096
        src_out[t]   = (int)(idx & (NN - 1));  // idx % 4096
        score_out[t] = __uint_as_float(bits);
    }
}

// ---------------------------------------------------------------------------
extern "C" void kernel_launch(void* const* d_in, const int* in_sizes, int n_in,
                              void* d_out, int out_size, void* d_ws, size_t ws_size,
                              hipStream_t stream)
{
    const float* ref = (const float*)d_in[0];
    const float* src = (const float*)d_in[1];

    uint8_t* ws = (uint8_t*)d_ws;
    size_t off = 0;
    float* S       = (float*)(ws + off); off += (size_t)NN * NN * sizeof(float);
    float* rowsum  = (float*)(ws + off); off += NN * sizeof(float);
    float* colsum  = (float*)(ws + off); off += NN * sizeof(float);
    float* rinv    = (float*)(ws + off); off += NN * sizeof(float);
    float* cinv    = (float*)(ws + off); off += NN * sizeof(float);
    unsigned int* ghist = (unsigned int*)(ws + off); off += HBINS * sizeof(unsigned int);
    unsigned int* ctrl  = (unsigned int*)(ws + off); off += 2 * sizeof(unsigned int);
    off = (off + 7) & ~(size_t)7;
    unsigned long long* cand = (unsigned long long*)(ws + off);
    off += CAND_MAX * sizeof(unsigned long long);

    // zero the accumulators (rowsum+colsum are contiguous) and the histogram
    hipMemsetAsync(rowsum, 0, 2 * NN * sizeof(float), stream);
    hipMemsetAsync(ghist, 0, HBINS * sizeof(unsigned int), stream);

    gemm_exp_kernel<<<dim3(NN / 128, NN / 128), 256,
                      128 * DD * sizeof(float), stream>>>(
        ref, src, S, rowsum, colsum);
    invert_kernel<<<NN / 256, 256, 0, stream>>>(rowsum, colsum, rinv, cinv);
    hist_kernel<<<1024, 256, HBINS * sizeof(unsigned int), stream>>>(
        S, rinv, cinv, ghist);
    scan_kernel<<<1, 256, HBINS * sizeof(unsigned int), stream>>>(ghist, ctrl);
    collect_kernel<<<1024, 256, 0, stream>>>(S, rinv, cinv, ctrl, cand);

    int* ref_out   = (int*)d_out;
    int* src_out   = ref_out + KSEL;
    float* sc_out  = (float*)d_out + 2 * KSEL;
    select_kernel<<<1, 1024, CAND_MAX * sizeof(unsigned long long), stream>>>(
        cand, ctrl, ref_out, src_out, sc_out);
}